// CAREGNN_62199716381202
// MI455X (gfx1250) — compile-verified
//
#include <hip/hip_runtime.h>
#include <hip/hip_bf16.h>
#include <math.h>

// ---------------------------------------------------------------------------
// CARE-GNN forward on MI455X (gfx1250, wave32).
//  - Edge aggregation: fp32 global atomics (L2-resident target, memory bound)
//  - Dense [N,128]x[128,64] GEMM: V_WMMA_F32_16X16X4_F32 (exact fp32 WMMA),
//    one wave per 16x64 strip: A frag loaded once, 4 independent acc chains.
//  - Tiny 2-column projections: plain VALU dot products
// ---------------------------------------------------------------------------

typedef __attribute__((ext_vector_type(2))) float v2f;
typedef __attribute__((ext_vector_type(8))) float v8f;

#define IN_DIM    128
#define HID_DIM   64

// ---------------- sim / final projection: X[N,D] @ W[D,2] + b -------------
__global__ void proj2_kernel(const float* __restrict__ X,
                             const float* __restrict__ W,
                             const float* __restrict__ b,
                             float* __restrict__ out,
                             int n, int D, int applyTanh) {
    int i = blockIdx.x * blockDim.x + threadIdx.x;
    if (i >= n) return;
    const float* x = X + (size_t)i * D;
    float s0 = 0.f, s1 = 0.f;
    for (int k = 0; k < D; k += 4) {
        float4 v = *(const float4*)(x + k);
        s0 = fmaf(v.x, W[(k+0)*2+0], s0); s1 = fmaf(v.x, W[(k+0)*2+1], s1);
        s0 = fmaf(v.y, W[(k+1)*2+0], s0); s1 = fmaf(v.y, W[(k+1)*2+1], s1);
        s0 = fmaf(v.z, W[(k+2)*2+0], s0); s1 = fmaf(v.z, W[(k+2)*2+1], s1);
        s0 = fmaf(v.w, W[(k+3)*2+0], s0); s1 = fmaf(v.w, W[(k+3)*2+1], s1);
    }
    s0 += b[0]; s1 += b[1];
    if (applyTanh) { s0 = tanhf(s0); s1 = tanhf(s1); }
    out[(size_t)i * 2 + 0] = s0;
    out[(size_t)i * 2 + 1] = s1;
}

// ---------------- edge scatter: aggsum[dst] += X[src], cnt[dst] += 1 ------
__global__ void scatter_kernel(const float* __restrict__ X,
                               const int* __restrict__ src,
                               const int* __restrict__ dst,
                               float* __restrict__ aggsum,
                               float* __restrict__ cnt,
                               int E, int D) {
    int t = blockIdx.x * blockDim.x + threadIdx.x;
    int nchunk = D >> 2;               // float4 chunks per row
    int e = t / nchunk;
    if (e >= E) return;
    int c = (t - e * nchunk) << 2;
    int s = src[e];
    int d = dst[e];
    const float4 v = *(const float4*)(X + (size_t)s * D + c);
    float* o = aggsum + (size_t)d * D + c;
    atomicAdd(o + 0, v.x);
    atomicAdd(o + 1, v.y);
    atomicAdd(o + 2, v.z);
    atomicAdd(o + 3, v.w);
    if (c == 0) atomicAdd(cnt + d, 1.0f);
}

// --------- hacc += p[pi] * tanh(aggsum / max(cnt,1)) ----------------------
__global__ void finish_mean_kernel(const float* __restrict__ aggsum,
                                   const float* __restrict__ cnt,
                                   const float* __restrict__ p, int pi,
                                   float* __restrict__ hacc,
                                   long total, int D) {
    long t = (long)blockIdx.x * blockDim.x + threadIdx.x;
    if (t >= total) return;
    long node = t / D;
    float c = fmaxf(cnt[node], 1.0f);
    hacc[t] += p[pi] * tanhf(aggsum[t] / c);
}

// --------- h = tanh(h + x) ------------------------------------------------
__global__ void residual_tanh_kernel(float* __restrict__ h,
                                     const float* __restrict__ x,
                                     long total) {
    long t = (long)blockIdx.x * blockDim.x + threadIdx.x;
    if (t >= total) return;
    h[t] = tanhf(h[t] + x[t]);
}

// --------- C[N,64] = A[N,128] @ B[128,64] + bias  (fp32 WMMA) -------------
// One wave per 16-row x 64-col strip: per K-step load the A frag ONCE and
// feed 4 independent 16x16 accumulator chains (4x A reuse, 4 parallel
// WMMA dependency chains to hide D->A/B RAW latency).
// A frag (16x4): lanes 0-15 hold row M=lane&15, K={k,k+1} with k=k0+2*(lane>>4).
// B frag (4x16): v0 = row k, v1 = row k+1, column = lane&15 (+16*j per tile).
// C/D: VGPR r -> row r + 8*(lane>>4), col = lane&15.
__global__ __launch_bounds__(256)
void gemm128x64_wmma_kernel(const float* __restrict__ A,
                            const float* __restrict__ B,
                            const float* __restrict__ bias,
                            float* __restrict__ C,
                            int nrows) {
    const int lane   = threadIdx.x;          // 0..31
    const int wave   = threadIdx.y;          // 0..7
    const int mtile  = blockIdx.x * 8 + wave;
    const int mt     = (nrows + 15) >> 4;
    if (mtile >= mt) return;

    const int half = lane >> 4;              // 0 or 1
    const int l15  = lane & 15;
    int rowA = mtile * 16 + l15;
    if (rowA >= nrows) rowA = nrows - 1;     // clamp OOB reads on edge tile

    const float* __restrict__ arow = A + (size_t)rowA * IN_DIM;

    v8f acc0 = {}, acc1 = {}, acc2 = {}, acc3 = {};
    #pragma unroll
    for (int k0 = 0; k0 < IN_DIM; k0 += 4) {
        const int k = k0 + 2 * half;
        v2f a; a.x = arow[k]; a.y = arow[k + 1];
        const float* __restrict__ brow0 = B + (size_t)k * HID_DIM + l15;
        const float* __restrict__ brow1 = brow0 + HID_DIM;
        v2f b0; b0.x = brow0[0];  b0.y = brow1[0];
        v2f b1; b1.x = brow0[16]; b1.y = brow1[16];
        v2f b2; b2.x = brow0[32]; b2.y = brow1[32];
        v2f b3; b3.x = brow0[48]; b3.y = brow1[48];
        acc0 = __builtin_amdgcn_wmma_f32_16x16x4_f32(false, a, false, b0,
                                                     (short)0, acc0, false, false);
        acc1 = __builtin_amdgcn_wmma_f32_16x16x4_f32(false, a, false, b1,
                                                     (short)0, acc1, false, false);
        acc2 = __builtin_amdgcn_wmma_f32_16x16x4_f32(false, a, false, b2,
                                                     (short)0, acc2, false, false);
        acc3 = __builtin_amdgcn_wmma_f32_16x16x4_f32(false, a, false, b3,
                                                     (short)0, acc3, false, false);
    }

    const float bv0 = bias[l15];
    const float bv1 = bias[l15 + 16];
    const float bv2 = bias[l15 + 32];
    const float bv3 = bias[l15 + 48];
    #pragma unroll
    for (int r = 0; r < 8; ++r) {
        int row = mtile * 16 + r + 8 * half;
        if (row < nrows) {
            float* crow = C + (size_t)row * HID_DIM + l15;
            crow[0]  = acc0[r] + bv0;
            crow[16] = acc1[r] + bv1;
            crow[32] = acc2[r] + bv2;
            crow[48] = acc3[r] + bv3;
        }
    }
}

// ---------------------------------------------------------------------------
extern "C" void kernel_launch(void* const* d_in, const int* in_sizes, int n_in,
                              void* d_out, int out_size, void* d_ws, size_t ws_size,
                              hipStream_t stream) {
    const float* feat = (const float*)d_in[0];
    const float* Wm   = (const float*)d_in[1];
    const float* bm   = (const float*)d_in[2];
    const float* W0   = (const float*)d_in[3];
    const float* b0   = (const float*)d_in[4];
    const float* W1   = (const float*)d_in[5];
    const float* b1   = (const float*)d_in[6];
    const float* p0   = (const float*)d_in[7];
    const float* p1   = (const float*)d_in[8];
    const int* srcs[3] = {(const int*)d_in[9],  (const int*)d_in[11], (const int*)d_in[13]};
    const int* dsts[3] = {(const int*)d_in[10], (const int*)d_in[12], (const int*)d_in[14]};

    const int N = in_sizes[0] / IN_DIM;   // 50000
    const int E = in_sizes[9];            // 500000

    float* out = (float*)d_out;                 // [N,2] layer-1 output (first)
    float* sim = out + (size_t)N * 2;           // [N,2] sim (second)

    // workspace layout (floats)
    float* ws    = (float*)d_ws;
    float* cnt   = ws;                  ws += N;
    float* aggs  = ws;                  ws += (size_t)N * IN_DIM;
    float* hacc  = ws;                  ws += (size_t)N * IN_DIM;   // layer-0 accum / h
    float* h1    = ws;                  ws += (size_t)N * HID_DIM;  // layer-0 GEMM out
    float* hacc1 = ws;                  ws += (size_t)N * HID_DIM;  // layer-1 accum / h2
    (void)ws_size; (void)n_in; (void)out_size;

    const int B256 = 256;

    // ---- sim = tanh(feat @ Wm + bm) ----
    proj2_kernel<<<(N + B256 - 1) / B256, B256, 0, stream>>>(
        feat, Wm, bm, sim, N, IN_DIM, 1);

    // ---- layer 0: per-etype mean-aggregate over feat (D=128) ----
    hipMemsetAsync(hacc, 0, (size_t)N * IN_DIM * sizeof(float), stream);
    {
        const long totE = (long)E * (IN_DIM / 4);
        const long totN = (long)N * IN_DIM;
        for (int i = 0; i < 3; ++i) {
            hipMemsetAsync(cnt,  0, (size_t)N * sizeof(float), stream);
            hipMemsetAsync(aggs, 0, (size_t)N * IN_DIM * sizeof(float), stream);
            scatter_kernel<<<(totE + B256 - 1) / B256, B256, 0, stream>>>(
                feat, srcs[i], dsts[i], aggs, cnt, E, IN_DIM);
            finish_mean_kernel<<<(totN + B256 - 1) / B256, B256, 0, stream>>>(
                aggs, cnt, p0, i, hacc, totN, IN_DIM);
        }
        residual_tanh_kernel<<<(totN + B256 - 1) / B256, B256, 0, stream>>>(
            hacc, feat, totN);
    }

    // ---- h1 = hacc @ W0 + b0  (fp32 WMMA GEMM, one wave per 16x64 strip) ----
    {
        const int mt = (N + 15) / 16;          // 3125 strips
        const int blocks = (mt + 7) / 8;
        gemm128x64_wmma_kernel<<<blocks, dim3(32, 8), 0, stream>>>(
            hacc, W0, b0, h1, N);
    }

    // ---- layer 1: per-etype mean-aggregate over h1 (D=64) ----
    hipMemsetAsync(hacc1, 0, (size_t)N * HID_DIM * sizeof(float), stream);
    {
        const long totE = (long)E * (HID_DIM / 4);
        const long totN = (long)N * HID_DIM;
        for (int i = 0; i < 3; ++i) {
            hipMemsetAsync(cnt,  0, (size_t)N * sizeof(float), stream);
            hipMemsetAsync(aggs, 0, (size_t)N * HID_DIM * sizeof(float), stream);
            scatter_kernel<<<(totE + B256 - 1) / B256, B256, 0, stream>>>(
                h1, srcs[i], dsts[i], aggs, cnt, E, HID_DIM);
            finish_mean_kernel<<<(totN + B256 - 1) / B256, B256, 0, stream>>>(
                aggs, cnt, p1, i, hacc1, totN, HID_DIM);
        }
        residual_tanh_kernel<<<(totN + B256 - 1) / B256, B256, 0, stream>>>(
            hacc1, h1, totN);
    }

    // ---- out = hacc1 @ W1 + b1 ----
    proj2_kernel<<<(N + B256 - 1) / B256, B256, 0, stream>>>(
        hacc1, W1, b1, out, N, HID_DIM, 0);
}